// TokenSelection_5454608466547
// MI455X (gfx1250) — compile-verified
//
#include <hip/hip_runtime.h>
#include <hip/hip_bf16.h>

// ---------------------------------------------------------------------------
// TokenSelection for gfx1250 (MI455X), wave32.
//   tokens:    [4, 16*196, 768] f32
//   attn_maps: [4, 16, 12, 12, 197, 197] f32
//   out:       [4, 16*49, 768] f32  ++  indices [4,16,49] i32 (flat concat)
// ---------------------------------------------------------------------------

typedef __attribute__((ext_vector_type(2))) float v2f;
typedef __attribute__((ext_vector_type(8))) float v8f;

#define NUM_BT   64          // B * NUM_FRAME
#define P_TOK    196
#define TOPK     49
#define EMB      768
#define MAP_STR  38809LL     // 197*197 elements between (l,h) maps
#define OUT_TOK_ELEMS (4LL * 16 * TOPK * EMB)   // 2,408,448 floats

// ---------------------------------------------------------------------------
// Stage 1: scores[bt, p] = sum over 48 attention rows of attn[..., 0, 1+p]
// One wave per (bt, 16-patch chunk); D = ones(16x4) x B(4x16) + C chained
// over 12 K-chunks via fp32 WMMA (exact fp32, matches reference top-k).
//
// No masking: WMMA columns are independent (D[:,n] depends only on B[:,n]),
// so out-of-range lanes load a clamped address and their columns are simply
// never stored. Row stride (155236 B) fits the 24-bit immediate offset, so
// all 24 loads share one base address.
// ---------------------------------------------------------------------------
__global__ void __launch_bounds__(32)
ts_scores_wmma(const float* __restrict__ attn, float* __restrict__ scores) {
    const int bt    = blockIdx.x;        // 0..63
    const int chunk = blockIdx.y;        // 0..12  (13 chunks of 16 patches)
    const int lane  = threadIdx.x;       // 0..31
    const int n     = lane & 15;         // column within chunk
    const int half  = lane >> 4;

    const int p  = chunk * 16 + n;
    const int pc = (p < P_TOK) ? p : (P_TOK - 1);   // clamp for address only

    // lane base: attn[b, t, layer 8, head 0, row 0, col 1+pc] + half*2 rows
    const float* lb = attn
        + (((long long)bt * 144 + 96) + (long long)(half * 2)) * MAP_STR
        + 1 + pc;

    const v2f a = {1.0f, 1.0f};          // A = ones(16x4)
    v8f acc = {};                        // C/D accumulator (f32, 8 VGPRs)

    #pragma unroll
    for (int k = 0; k < 12; ++k) {       // rows k*4 + half*2 + {0,1}
        v2f bv;
        bv.x = lb[(long long)(k * 4    ) * MAP_STR];   // imm offsets <= 7.0MB
        bv.y = lb[(long long)(k * 4 + 1) * MAP_STR];
        acc = __builtin_amdgcn_wmma_f32_16x16x4_f32(
                  false, a, false, bv, (short)0, acc, false, false);
    }

    // D row M=0 lives in acc[0] of lanes 0..15 (column N = lane).
    if (half == 0 && p < P_TOK) scores[bt * P_TOK + p] = acc[0];
}

// ---------------------------------------------------------------------------
// Stage 2: top-49 by rank. rank[p] = #{q: s[q]>s[p] or (s[q]==s[p] && q<p)}.
// Ranks 0..48 reproduce jax top_k descending order with stable tie-break.
// ---------------------------------------------------------------------------
__global__ void __launch_bounds__(256)
ts_topk(const float* __restrict__ scores, int* __restrict__ indices) {
    __shared__ float s[P_TOK];
    const int bt  = blockIdx.x;
    const int tid = threadIdx.x;

    if (tid < P_TOK) s[tid] = scores[bt * P_TOK + tid];
    __syncthreads();

    if (tid < P_TOK) {
        const float v = s[tid];
        int rank = 0;
        #pragma unroll 4
        for (int q = 0; q < P_TOK; ++q) {
            const float w = s[q];
            rank += (w > v) || (w == v && q < tid);
        }
        if (rank < TOPK) indices[bt * TOPK + rank] = tid;
    }
}

// ---------------------------------------------------------------------------
// Stage 3: gather selected token rows. One block per output row; 192 lanes
// move 768 floats as float4 -> global_load_b128 / global_store_b128.
// ---------------------------------------------------------------------------
__global__ void __launch_bounds__(192)
ts_gather(const float* __restrict__ tokens, const int* __restrict__ indices,
          float* __restrict__ out) {
    const int row = blockIdx.x;              // (bt)*49 + k, 0..3135
    const int bt  = row / TOPK;
    const int idx = indices[row];

    const float4* __restrict__ src =
        (const float4*)(tokens + ((long long)bt * P_TOK + idx) * EMB);
    float4* __restrict__ dst = (float4*)(out + (long long)row * EMB);
    dst[threadIdx.x] = src[threadIdx.x];     // 192 * 16B = 3072B per row
}

// ---------------------------------------------------------------------------
extern "C" void kernel_launch(void* const* d_in, const int* in_sizes, int n_in,
                              void* d_out, int out_size, void* d_ws, size_t ws_size,
                              hipStream_t stream) {
    const float* tokens = (const float*)d_in[0];
    const float* attn   = (const float*)d_in[1];

    float* out_tok = (float*)d_out;
    int*   out_idx = (int*)d_out + OUT_TOK_ELEMS;   // indices region (i32 bits)

    float* scores = (float*)d_ws;                   // 64*196 floats = 50 KB

    dim3 g1(NUM_BT, 13);                             // 13 chunks of 16 patches
    ts_scores_wmma<<<g1, 32, 0, stream>>>(attn, scores);
    ts_topk<<<NUM_BT, 256, 0, stream>>>(scores, out_idx);
    ts_gather<<<NUM_BT * TOPK, 192, 0, stream>>>(tokens, out_idx, out_tok);
}